// SSIMLoss_10617159155794
// MI455X (gfx1250) — compile-verified
//
#include <hip/hip_runtime.h>

// ---------------------------------------------------------------------------
// SSIM loss, fused single-pass:  16x3x512x512 fp32, 7x7 box windows (valid),
// S-map mean, out = 1 - mean.  Bandwidth-bound (~100 MB in / 23.3 TB/s).
// Horizontal 7-tap sums done on the matrix pipe via V_WMMA_F32_16X16X4_F32:
//   Out(16x16) = ColSums(16x24) x BandMatrix(24x16),  band built in registers.
// ---------------------------------------------------------------------------

typedef float v2f __attribute__((ext_vector_type(2)));
typedef float v8f __attribute__((ext_vector_type(8)));

#if defined(__has_builtin)
#if __has_builtin(__builtin_amdgcn_wmma_f32_16x16x4_f32)
#define USE_WMMA_F32 1
#endif
#endif

namespace {
constexpr int H = 512, W = 512;
constexpr int OH = 506, OW = 506;          // valid region (512 - 6)
constexpr int PLANES = 16 * 3;             // 48 (B*C) planes, plane-major layout
constexpr int TILE_H = 16, TILE_W = 64;    // per-block output tile
constexpr int TILES_X = 8, STRIPS = 32;    // 8*64 = 512, 32*16 = 512 (masked)
constexpr int NBLOCKS = PLANES * STRIPS * TILES_X;  // 12288
constexpr int IN_H = TILE_H + 6;           // 22 input rows
constexpr int IN_W = TILE_W + 6;           // 70 input cols
constexpr int XSTRIDE = IN_W + 1;          // 71, LDS pad
constexpr int CS_COLS = 72;                // 70 used + 2 zero-weight pad (K pad to 24)
constexpr int CS_RSTR = 17;                // 16 rows + pad -> conflict-free fetch
}

__global__ __launch_bounds__(128) void ssim_partial_kernel(
    const float* __restrict__ pred, const float* __restrict__ targ,
    float* __restrict__ partials) {
  __shared__ float xt[IN_H * XSTRIDE];
  __shared__ float yt[IN_H * XSTRIDE];
  __shared__ float cs[5 * CS_COLS * CS_RSTR];  // [stat][col][row], transposed
  __shared__ float red[128];

  const int tid = threadIdx.x;
  const int bid = blockIdx.x;
  const int plane = bid >> 8;        // / 256 blocks-per-plane
  const int rem = bid & 255;
  const int strip = rem >> 3;        // 0..31
  const int tx = rem & 7;            // 0..7
  const size_t pbase = (size_t)plane * (size_t)(H * W);
  const int gr0 = strip * TILE_H;
  const int gc0 = tx * TILE_W;

  // ---- stage clipped input tiles (22 x 70) into LDS, coalesced ----
  for (int idx = tid; idx < IN_H * IN_W; idx += 128) {
    const int r = idx / IN_W, c = idx - r * IN_W;
    const int gr = min(gr0 + r, H - 1);
    const int gc = min(gc0 + c, W - 1);
    const size_t off = pbase + (size_t)gr * W + gc;
    float xv = pred[off]; xv = fminf(fmaxf(xv, 0.0f), 1.0f);
    float yv = targ[off]; yv = fminf(fmaxf(yv, 0.0f), 1.0f);
    xt[r * XSTRIDE + c] = xv;
    yt[r * XSTRIDE + c] = yv;
  }
  // speculative prefetch of the next row-strip (global_prefetch_b8)
  if (tid < 32) {
    const int pr = min(gr0 + IN_H + (tid & 15), H - 1);
    const float* p = (tid < 16) ? pred : targ;
    __builtin_prefetch(&p[pbase + (size_t)pr * W + gc0], 0, 0);
  }
  __syncthreads();

  // ---- vertical 7-tap sums for 5 stats; store transposed [col][row] ----
  for (int idx = tid; idx < CS_COLS * TILE_H; idx += 128) {
    const int c = idx >> 4;           // 0..71
    const int m = idx & 15;           // output row in strip
    const int csrc = min(c, IN_W - 1);  // pad cols read clamped (zero weight later)
    float sx = 0.f, sy = 0.f, sxx = 0.f, syy = 0.f, sxy = 0.f;
#pragma unroll
    for (int k = 0; k < 7; ++k) {
      const float xv = xt[(m + k) * XSTRIDE + csrc];
      const float yv = yt[(m + k) * XSTRIDE + csrc];
      sx += xv; sy += yv;
      sxx = fmaf(xv, xv, sxx);
      syy = fmaf(yv, yv, syy);
      sxy = fmaf(xv, yv, sxy);
    }
    const int o = c * CS_RSTR + m;
    cs[0 * CS_COLS * CS_RSTR + o] = sx;
    cs[1 * CS_COLS * CS_RSTR + o] = sy;
    cs[2 * CS_COLS * CS_RSTR + o] = sxx;
    cs[3 * CS_COLS * CS_RSTR + o] = syy;
    cs[4 * CS_COLS * CS_RSTR + o] = sxy;
  }
  __syncthreads();

  // ---- horizontal 7-tap sums: Out(16x16) = CS(16x24) x Band(24x16) ----
  const int lane = tid & 31;
  const int wv = tid >> 5;            // wave 0..3 -> 16-col slab
  const int lm = lane & 15;           // A row M and B col N
  const int hi = lane >> 4;           // half-wave: K split
  const int wcol0 = wv * 16;

  v8f acc[5];
#pragma unroll
  for (int s = 0; s < 5; ++s)
#pragma unroll
    for (int j = 0; j < 8; ++j) acc[s][j] = 0.0f;

#if defined(USE_WMMA_F32)
#pragma unroll
  for (int t = 0; t < 6; ++t) {
    const int k0 = 4 * t + 2 * hi;    // per-lane K indices {k0, k0+1}
    // band matrix B[k][n] = 1 iff n <= k <= n+6 (cols 22,23 naturally zero)
    v2f bv;
    bv.x = (k0 >= lm && k0 <= lm + 6) ? 1.0f : 0.0f;
    bv.y = (k0 + 1 >= lm && k0 + 1 <= lm + 6) ? 1.0f : 0.0f;
#pragma unroll
    for (int s = 0; s < 5; ++s) {
      const float* csb = &cs[(s * CS_COLS + wcol0 + k0) * CS_RSTR + lm];
      v2f av;
      av.x = csb[0];
      av.y = csb[CS_RSTR];
      acc[s] = __builtin_amdgcn_wmma_f32_16x16x4_f32(
          false, av, false, bv, (short)0, acc[s], false, false);
    }
  }
#else
  // VALU fallback, numerically identical
#pragma unroll
  for (int j = 0; j < 8; ++j) {
    const int row = j + 8 * hi;
#pragma unroll
    for (int s = 0; s < 5; ++s) {
      float v = 0.f;
#pragma unroll
      for (int d = 0; d < 7; ++d)
        v += cs[(s * CS_COLS + wcol0 + lm + d) * CS_RSTR + row];
      acc[s][j] = v;
    }
  }
#endif

  // ---- fused SSIM map + masked accumulation ----
  constexpr float INV_NP = 1.0f / 49.0f;
  constexpr float COVN = 49.0f / 48.0f;
  constexpr float C1 = 1.0e-4f;   // (0.01)^2
  constexpr float C2 = 9.0e-4f;   // (0.03)^2
  float tsum = 0.0f;
#pragma unroll
  for (int j = 0; j < 8; ++j) {
    const int orow = gr0 + j + 8 * hi;
    const int ocol = gc0 + wcol0 + lm;
    const float ux = acc[0][j] * INV_NP;
    const float uy = acc[1][j] * INV_NP;
    const float uxx = acc[2][j] * INV_NP;
    const float uyy = acc[3][j] * INV_NP;
    const float uxy = acc[4][j] * INV_NP;
    const float vx = COVN * (uxx - ux * ux);
    const float vy = COVN * (uyy - uy * uy);
    const float vxy = COVN * (uxy - ux * uy);
    const float num = (2.0f * ux * uy + C1) * (2.0f * vxy + C2);
    const float den = (ux * ux + uy * uy + C1) * (vx + vy + C2);
    const float S = num / den;
    if (orow < OH && ocol < OW) tsum += S;
  }

  // ---- deterministic in-block tree reduction ----
  red[tid] = tsum;
  __syncthreads();
#pragma unroll
  for (int off = 64; off > 0; off >>= 1) {
    if (tid < off) red[tid] += red[tid + off];
    __syncthreads();
  }
  if (tid == 0) partials[bid] = red[0];
}

__global__ __launch_bounds__(256) void ssim_finalize_kernel(
    const float* __restrict__ partials, int n, float* __restrict__ out) {
  __shared__ double red[256];
  const int tid = threadIdx.x;
  double s = 0.0;
  for (int i = tid; i < n; i += 256) s += (double)partials[i];
  red[tid] = s;
  __syncthreads();
#pragma unroll
  for (int off = 128; off > 0; off >>= 1) {
    if (tid < off) red[tid] += red[tid + off];
    __syncthreads();
  }
  if (tid == 0) {
    const double N = (double)PLANES * (double)OH * (double)OW;  // 12,289,728
    out[0] = (float)(1.0 - red[0] / N);
  }
}

extern "C" void kernel_launch(void* const* d_in, const int* in_sizes, int n_in,
                              void* d_out, int out_size, void* d_ws, size_t ws_size,
                              hipStream_t stream) {
  (void)in_sizes; (void)n_in; (void)out_size; (void)ws_size;
  const float* pred = (const float*)d_in[0];
  const float* targ = (const float*)d_in[1];
  float* partials = (float*)d_ws;  // NBLOCKS floats (~48 KB)

  ssim_partial_kernel<<<dim3(NBLOCKS), dim3(128), 0, stream>>>(pred, targ, partials);
  ssim_finalize_kernel<<<dim3(1), dim3(256), 0, stream>>>(partials, NBLOCKS,
                                                          (float*)d_out);
}